// SoftFAPELoss_37168646980031
// MI455X (gfx1250) — compile-verified
//
#include <hip/hip_runtime.h>
#include <math.h>

typedef __attribute__((ext_vector_type(2))) float v2f;
typedef __attribute__((ext_vector_type(8))) float v8f;

#define BB 8
#define NN 4096
#define MM 4096
#define CHUNK 2048                 // true points staged in LDS per step (32 KB)
#define NCHUNK (MM / CHUNK)
#define TILES_PER_CHUNK (CHUNK / 16)

// ---------------------------------------------------------------------------
// Prep: rigid transform + pack augmented float4 panels.
//   P[b,n] = (-2x', -2y', -2z', |p'|^2)   -> A rows use (.x,.y)/( .z, 1.0 )
//                                            and .w feeds the C (row bias)
//   Q[b,m] = ( x',   y',   z',  |t'|^2)   -> B cols use (.x,.y)/( .z, .w )
// D = A*B + C = -2 p.t + |t|^2 + |p|^2 = dist^2, exactly.
// Also zeroes the scalar output (stream-ordered before the main kernel).
// ---------------------------------------------------------------------------
__global__ __launch_bounds__(256) void fape_prep(
    const float* __restrict__ Xp, const float* __restrict__ Xt,
    const float* __restrict__ Rp, const float* __restrict__ tp,
    const float* __restrict__ Rt, const float* __restrict__ tt,
    float4* __restrict__ P, float4* __restrict__ Q, float* __restrict__ out)
{
    int gid = blockIdx.x * blockDim.x + threadIdx.x;
    if (gid == 0) out[0] = 0.0f;

    if (gid < BB * NN) {                      // pred points
        int b = gid / NN;
        const float* x = Xp + (size_t)gid * 3;
        const float* R = Rp + b * 9;
        const float* t = tp + b * 3;
        float x0 = x[0], x1 = x[1], x2 = x[2];
        float y0 = R[0]*x0 + R[1]*x1 + R[2]*x2 + t[0];
        float y1 = R[3]*x0 + R[4]*x1 + R[5]*x2 + t[1];
        float y2 = R[6]*x0 + R[7]*x1 + R[8]*x2 + t[2];
        float sq = y0*y0 + y1*y1 + y2*y2;
        P[gid] = make_float4(-2.0f*y0, -2.0f*y1, -2.0f*y2, sq);
    } else if (gid < 2 * BB * NN) {           // true points (N == M)
        int g = gid - BB * NN;
        int b = g / MM;
        const float* x = Xt + (size_t)g * 3;
        const float* R = Rt + b * 9;
        const float* t = tt + b * 3;
        float x0 = x[0], x1 = x[1], x2 = x[2];
        float y0 = R[0]*x0 + R[1]*x1 + R[2]*x2 + t[0];
        float y1 = R[3]*x0 + R[4]*x1 + R[5]*x2 + t[1];
        float y2 = R[6]*x0 + R[7]*x1 + R[8]*x2 + t[2];
        float sq = y0*y0 + y1*y1 + y2*y2;
        Q[g] = make_float4(y0, y1, y2, sq);
    }
}

__device__ __forceinline__ float clamp0(float x) {
    // single v_med3_f32: clamp to [0, +inf) without fmaxf's canonicalize pair
    return __builtin_amdgcn_fmed3f(x, 0.0f, __builtin_inff());
}

// ---------------------------------------------------------------------------
// Main: per wave, one 16-row tile of pred points; sweep all M true points in
// 16-col tiles via V_WMMA_F32_16X16X4_F32 (D = dist_sq directly).
// Two passes: (1) row-min of dist, (2) exp + weighted accumulation.
// ---------------------------------------------------------------------------
__global__ __launch_bounds__(256) void fape_main(
    const float4* __restrict__ P, const float4* __restrict__ Q,
    const float* __restrict__ temp, float* __restrict__ out)
{
    __shared__ float4 qsh[CHUNK];             // 32 KB

    const int b      = blockIdx.x >> 5;       // 32 groups per batch
    const int grp    = blockIdx.x & 31;
    const int tid    = threadIdx.x;
    const int wave   = tid >> 5;
    const int lane   = tid & 31;
    const int lane16 = lane & 15;
    const int half   = lane >> 4;

    const int n0  = grp * 128 + wave * 16;
    const int row = n0 + lane16;

    // ---- A fragment: one uniform 8-byte load, one cndmask fixup ----
    const v2f* P2 = (const v2f*)P;
    v2f a = P2[(size_t)(b * NN + row) * 2 + half];   // (−2x,−2y) or (−2z, sq)
    float sqp = a.y;                                  // high half holds sq here
    sqp = __shfl(sqp, lane16 + 16, 32);               // sq_p of own row, all lanes
    a.y = half ? 1.0f : a.y;                          // A K=3 must be 1.0

    // ---- C fragment (loop-invariant): C[v][lane] = sq_p[row v + 8*half] ----
    v8f cc;
#pragma unroll
    for (int v = 0; v < 8; ++v)
        cc[v] = __shfl(sqp, v + 8 * half, 32);

    const float T     = temp[0];
    const float scale = 1.4426950408889634f / T;      // log2(e)/T

    const v2f* q2 = (const v2f*)qsh;
    const int  bfi = lane16 * 2 + half;               // v2f index within a tile

    float mn[8];
#pragma unroll
    for (int v = 0; v < 8; ++v) mn[v] = 3.0e38f;

    // ----- pass 1: row minimum of dist_sq (no clamp needed inside) -----
    for (int c = 0; c < NCHUNK; ++c) {
        __syncthreads();
        for (int i = tid; i < CHUNK; i += 256)
            qsh[i] = Q[b * MM + c * CHUNK + i];
        __syncthreads();

        for (int mt = 0; mt < TILES_PER_CHUNK; ++mt) {
            v2f bf = q2[mt * 32 + bfi];               // one ds_load_b64
            v8f d = __builtin_amdgcn_wmma_f32_16x16x4_f32(
                false, a, false, bf, (short)0, cc, false, false);
#pragma unroll
            for (int v = 0; v < 8; ++v)
                mn[v] = fminf(mn[v], d[v]);
        }
    }

    // reduce min across the 16 lanes holding each row (xor 1,2,4,8 stays in half)
#pragma unroll
    for (int v = 0; v < 8; ++v) {
#pragma unroll
        for (int m = 1; m <= 8; m <<= 1)
            mn[v] = fminf(mn[v], __shfl_xor(mn[v], m, 32));
        mn[v] = clamp0(mn[v]);                        // clamp commutes with min
    }

    // pre-scale row max: exp arg = fma(dv, -scale, mns)
    float mns[8];
#pragma unroll
    for (int v = 0; v < 8; ++v) mns[v] = mn[v] * scale;

    // ----- pass 2: softmax-weighted distance -----
    float se[8], sw[8];
#pragma unroll
    for (int v = 0; v < 8; ++v) { se[v] = 0.0f; sw[v] = 0.0f; }

    for (int c = 0; c < NCHUNK; ++c) {
        __syncthreads();
        for (int i = tid; i < CHUNK; i += 256)
            qsh[i] = Q[b * MM + c * CHUNK + i];
        __syncthreads();

        for (int mt = 0; mt < TILES_PER_CHUNK; ++mt) {
            v2f bf = q2[mt * 32 + bfi];
            v8f d = __builtin_amdgcn_wmma_f32_16x16x4_f32(
                false, a, false, bf, (short)0, cc, false, false);
#pragma unroll
            for (int v = 0; v < 8; ++v) {
                float dv = clamp0(d[v]);                            // v_med3
                float e  = __builtin_amdgcn_exp2f(
                               __builtin_fmaf(dv, -scale, mns[v])); // fma + exp
                se[v] += e;
                sw[v] = __builtin_fmaf(dv, e, sw[v]);
            }
        }
    }

    // reduce sums across the 16 lanes of each half
#pragma unroll
    for (int v = 0; v < 8; ++v) {
#pragma unroll
        for (int m = 1; m <= 8; m <<= 1) {
            se[v] += __shfl_xor(se[v], m, 32);
            sw[v] += __shfl_xor(sw[v], m, 32);
        }
    }

    // lane 0 owns rows 0..7 of the tile, lane 16 rows 8..15
    if (lane16 == 0) {
        float acc = 0.0f;
#pragma unroll
        for (int v = 0; v < 8; ++v) acc += sw[v] / se[v];
        atomicAdd(out, acc * (1.0f / (float)(BB * NN)));
    }
}

// ---------------------------------------------------------------------------
extern "C" void kernel_launch(void* const* d_in, const int* in_sizes, int n_in,
                              void* d_out, int out_size, void* d_ws, size_t ws_size,
                              hipStream_t stream)
{
    const float* Xp = (const float*)d_in[0];
    const float* Xt = (const float*)d_in[1];
    const float* Rp = (const float*)d_in[2];
    const float* tp = (const float*)d_in[3];
    const float* Rt = (const float*)d_in[4];
    const float* tt = (const float*)d_in[5];
    const float* temperature = (const float*)d_in[6];
    float* out = (float*)d_out;

    float4* P = (float4*)d_ws;                 // B*N float4 = 512 KB
    float4* Q = P + (size_t)BB * NN;           // B*M float4 = 512 KB

    // prep: 2*B*N = 65536 threads
    fape_prep<<<(2 * BB * NN) / 256, 256, 0, stream>>>(Xp, Xt, Rp, tp, Rt, tt, P, Q, out);

    // main: B * (N/128) = 256 blocks of 256 threads (8 waves, 16 rows each)
    fape_main<<<BB * 32, 256, 0, stream>>>(P, Q, temperature, out);
}